// GAT_14465449853560
// MI455X (gfx1250) — compile-verified
//
#include <hip/hip_runtime.h>
#include <hip/hip_bf16.h>

// ---------------------------------------------------------------------------
// SuperGAT (2 layers, MX attention) forward for MI455X / gfx1250.
// GEMMs use V_WMMA_F32_16X16X4_F32 (exact f32). Edge phase is 3-pass
// segment-softmax with global f32 atomics (L2-resident xt: 25.6MB << 192MB L2).
// ---------------------------------------------------------------------------

#define NNODES  50000
#define EEDGES  800000
#define ETOT    (EEDGES + NNODES)   // 850000 (self-loops appended)
#define FDIM    128
#define NHEADS  8
#define CDIM    16

typedef __attribute__((ext_vector_type(2))) float v2f;
typedef __attribute__((ext_vector_type(8))) float v8f;

// ---- order-preserving float<->uint encoding for atomic max ----------------
__device__ __forceinline__ unsigned fenc(float f) {
    unsigned u = __float_as_uint(f);
    return (u & 0x80000000u) ? ~u : (u | 0x80000000u);
}
__device__ __forceinline__ float fdec(unsigned u) {
    u = (u & 0x80000000u) ? (u ^ 0x80000000u) : ~u;
    return __uint_as_float(u);
}

__device__ __forceinline__ void edge_ends(const int* __restrict__ ei, int e,
                                          int& s, int& d) {
    if (e < EEDGES) { s = ei[e]; d = ei[EEDGES + e]; }
    else            { s = e - EEDGES; d = s; }
}

// ---------------------------------------------------------------------------
// GEMM: Out[N,128] = A[N,128] @ W[128,128] via v_wmma_f32_16x16x4_f32.
// Block = 256 threads (8 waves). Block owns a 16-row tile; wave w owns the
// 16-col tile [16w, 16w+16). A-tile staged in LDS (stride 132, conflict-free).
// ---------------------------------------------------------------------------
__global__ void gat_gemm128_wmma(const float* __restrict__ Ain,
                                 const float* __restrict__ Wmat,
                                 float* __restrict__ Out) {
    __shared__ float As[16 * 132];
    const int tileBase = blockIdx.x * 16;        // N = 50000 = 3125*16 exact
    const int t = threadIdx.x;

    // Stage 16x128 A-tile into LDS (2 float4 per thread, coalesced).
    const float4* A4 = (const float4*)(Ain + (size_t)tileBase * FDIM);
    #pragma unroll
    for (int j = 0; j < 2; ++j) {
        int gi = j * 256 + t;                    // float4 index in 16x32 tile
        int r  = gi >> 5;
        int c4 = gi & 31;
        float4 v = A4[r * 32 + c4];
        *(float4*)&As[r * 132 + c4 * 4] = v;     // (r*132 + 4c)*4B is 16B-aligned
    }
    __syncthreads();

    const int wave = t >> 5;
    const int lane = t & 31;
    const int m    = lane & 15;
    const int half = lane >> 4;                  // 0: K pair {0,1}, 1: {2,3}
    const int col  = wave * 16 + m;

    v8f acc = {};
    #pragma unroll 4
    for (int kk = 0; kk < 32; ++kk) {
        const int k = kk * 4 + half * 2;
        v2f a, b;
        // A fragment (16x4 MxK): lane m, VGPR0=K=k, VGPR1=K=k+1
        a.x = As[m * 132 + k];
        a.y = As[m * 132 + k + 1];
        // B fragment (4x16 KxN): rows striped across lanes within a VGPR
        b.x = Wmat[(size_t)k * FDIM + col];
        b.y = Wmat[(size_t)(k + 1) * FDIM + col];
        acc = __builtin_amdgcn_wmma_f32_16x16x4_f32(
            /*neg_a=*/false, a, /*neg_b=*/false, b,
            /*c_mod=*/(short)0, acc, /*reuse_a=*/false, /*reuse_b=*/false);
    }

    // C/D 16x16 f32 layout: VGPR i -> M=i (lanes 0-15) / M=i+8 (lanes 16-31)
    const int mbase = half * 8;
    #pragma unroll
    for (int i = 0; i < 8; ++i)
        Out[(size_t)(tileBase + mbase + i) * FDIM + wave * 16 + m] = acc[i];
}

// ---------------------------------------------------------------------------
__global__ void gat_zero(float* __restrict__ p, int n) {
    int g = blockIdx.x * blockDim.x + threadIdx.x;
    if (g < n) p[g] = 0.0f;                      // also 0u for the amax region
}

// Pass A: alpha = leaky_relu((aL.xj + aR.xi) * sigmoid(xi.xj)); amax via atomicMax
__global__ void gat_attn_alpha(const float* __restrict__ xt,
                               const float* __restrict__ attL,
                               const float* __restrict__ attR,
                               const int* __restrict__ ei,
                               float* __restrict__ alpha,
                               unsigned* __restrict__ amax) {
    int gid = blockIdx.x * blockDim.x + threadIdx.x;
    if (gid >= ETOT * NHEADS) return;
    int e = gid >> 3, hd = gid & 7;
    int s, d; edge_ends(ei, e, s, d);

    const float4* xj = (const float4*)(xt + (size_t)s * FDIM + hd * CDIM);
    const float4* xi = (const float4*)(xt + (size_t)d * FDIM + hd * CDIM);
    const float4* al = (const float4*)(attL + hd * CDIM);
    const float4* ar = (const float4*)(attR + hd * CDIM);

    float logit = 0.f, a = 0.f;
    #pragma unroll
    for (int q = 0; q < 4; ++q) {
        float4 vj = xj[q], vi = xi[q], l = al[q], r = ar[q];
        logit += vi.x * vj.x + vi.y * vj.y + vi.z * vj.z + vi.w * vj.w;
        a += l.x * vj.x + l.y * vj.y + l.z * vj.z + l.w * vj.w;
        a += r.x * vi.x + r.y * vi.y + r.z * vi.z + r.w * vi.w;
    }
    float sg = 1.0f / (1.0f + expf(-logit));
    float v  = a * sg;
    v = (v > 0.0f) ? v : 0.2f * v;               // leaky_relu slope 0.2
    alpha[gid] = v;
    atomicMax(&amax[(size_t)d * NHEADS + hd], fenc(v));
}

// Pass B: ex = exp(alpha - amax[dst]); denom[dst] += ex (alpha overwritten)
__global__ void gat_attn_exp(float* __restrict__ alpha,
                             const unsigned* __restrict__ amax,
                             float* __restrict__ denom,
                             const int* __restrict__ ei) {
    int gid = blockIdx.x * blockDim.x + threadIdx.x;
    if (gid >= ETOT * NHEADS) return;
    int e = gid >> 3, hd = gid & 7;
    int s, d; edge_ends(ei, e, s, d); (void)s;
    float m  = fdec(amax[(size_t)d * NHEADS + hd]);
    float ex = expf(alpha[gid] - m);
    alpha[gid] = ex;
    atomicAdd(&denom[(size_t)d * NHEADS + hd], ex);
}

// Pass C: agg[dst] += x_j * (ex / (denom[dst] + 1e-16))
__global__ void gat_attn_aggregate(const float* __restrict__ alpha,
                                   const float* __restrict__ denom,
                                   const float* __restrict__ xt,
                                   float* __restrict__ agg,
                                   const int* __restrict__ ei) {
    int gid = blockIdx.x * blockDim.x + threadIdx.x;
    if (gid >= ETOT * NHEADS) return;
    int e = gid >> 3, hd = gid & 7;
    int s, d; edge_ends(ei, e, s, d);
    float attn = alpha[gid] / (denom[(size_t)d * NHEADS + hd] + 1e-16f);
    const float* xj = xt + (size_t)s * FDIM + hd * CDIM;
    float* o = agg + (size_t)d * FDIM + hd * CDIM;
    #pragma unroll
    for (int c = 0; c < CDIM; ++c)
        atomicAdd(&o[c], xj[c] * attn);
}

// bias + exact GELU
__global__ void gat_bias_gelu(const float* __restrict__ agg,
                              const float* __restrict__ bias,
                              float* __restrict__ out) {
    int g = blockIdx.x * blockDim.x + threadIdx.x;
    if (g >= NNODES * FDIM) return;
    float v = agg[g] + bias[g & (FDIM - 1)];
    out[g] = 0.5f * v * (1.0f + erff(v * 0.70710678118654752f));
}

// final FC: y[n] = h[n,:] . fcW + fcb
__global__ void gat_fc(const float* __restrict__ h,
                       const float* __restrict__ w,
                       const float* __restrict__ b0,
                       float* __restrict__ out) {
    int n = blockIdx.x * blockDim.x + threadIdx.x;
    if (n >= NNODES) return;
    const float4* h4 = (const float4*)(h + (size_t)n * FDIM);
    const float4* w4 = (const float4*)w;
    float s = 0.f;
    #pragma unroll 8
    for (int i = 0; i < FDIM / 4; ++i) {
        float4 a = h4[i], c = w4[i];
        s += a.x * c.x + a.y * c.y + a.z * c.z + a.w * c.w;
    }
    out[n] = s + b0[0];
}

// ---------------------------------------------------------------------------
extern "C" void kernel_launch(void* const* d_in, const int* in_sizes, int n_in,
                              void* d_out, int out_size, void* d_ws, size_t ws_size,
                              hipStream_t stream) {
    (void)in_sizes; (void)n_in; (void)out_size; (void)ws_size;
    const float* x   = (const float*)d_in[0];
    const float* W1  = (const float*)d_in[1];
    const float* b1  = (const float*)d_in[2];
    const float* al1 = (const float*)d_in[3];
    const float* ar1 = (const float*)d_in[4];
    const float* W2  = (const float*)d_in[5];
    const float* b2  = (const float*)d_in[6];
    const float* al2 = (const float*)d_in[7];
    const float* ar2 = (const float*)d_in[8];
    const float* fcW = (const float*)d_in[9];
    const float* fcb = (const float*)d_in[10];
    const int*   ei  = (const int*)d_in[11];

    // workspace layout (floats): xt | hbuf | alpha | agg | denom | amax
    float*    ws    = (float*)d_ws;
    float*    xt    = ws;                               // N*128
    float*    hbuf  = xt    + (size_t)NNODES * FDIM;    // N*128
    float*    alpha = hbuf  + (size_t)NNODES * FDIM;    // ETOT*8
    float*    agg   = alpha + (size_t)ETOT * NHEADS;    // N*128
    float*    denom = agg   + (size_t)NNODES * FDIM;    // N*8
    unsigned* amax  = (unsigned*)(denom + (size_t)NNODES * NHEADS); // N*8

    const int ZN = NNODES * (FDIM + 2 * NHEADS);        // agg+denom+amax contiguous
    const int EG = (ETOT * NHEADS + 255) / 256;
    const int NG = (NNODES * FDIM + 255) / 256;
    const dim3 B(256);

    // ---- layer 1 ----
    gat_gemm128_wmma<<<NNODES / 16, B, 0, stream>>>(x, W1, xt);
    gat_zero<<<(ZN + 255) / 256, B, 0, stream>>>(agg, ZN);
    gat_attn_alpha<<<EG, B, 0, stream>>>(xt, al1, ar1, ei, alpha, amax);
    gat_attn_exp<<<EG, B, 0, stream>>>(alpha, amax, denom, ei);
    gat_attn_aggregate<<<EG, B, 0, stream>>>(alpha, denom, xt, agg, ei);
    gat_bias_gelu<<<NG, B, 0, stream>>>(agg, b1, hbuf);

    // ---- layer 2 ----
    gat_gemm128_wmma<<<NNODES / 16, B, 0, stream>>>(hbuf, W2, xt);
    gat_zero<<<(ZN + 255) / 256, B, 0, stream>>>(agg, ZN);
    gat_attn_alpha<<<EG, B, 0, stream>>>(xt, al2, ar2, ei, alpha, amax);
    gat_attn_exp<<<EG, B, 0, stream>>>(alpha, amax, denom, ei);
    gat_attn_aggregate<<<EG, B, 0, stream>>>(alpha, denom, xt, agg, ei);
    gat_bias_gelu<<<NG, B, 0, stream>>>(agg, b2, hbuf);

    // ---- final FC ----
    gat_fc<<<(NNODES + 255) / 256, B, 0, stream>>>(hbuf, fcW, fcb, (float*)d_out);
}